// TaylorForward_36979668419222
// MI455X (gfx1250) — compile-verified
//
#include <hip/hip_runtime.h>
#include <math.h>

// ---------------------------------------------------------------------------
// Fused  proj = P @ theta_dir  (WMMA f32 16x16x4, K=64)  +  Taylor/exp epilogue
// Shapes: B=2, DIRS=64, C=630, H=W=128 (HW=16384)
//   theta_global : (B,4,H,W)      f32
//   theta_dir    : (B,64,6,H,W)   f32
//   t_vec, b_vec : (C,)           f32
//   P            : (C,64)         f32
//   out          : (B,C,H,W)      f32
// proj (B,6,C,H,W) = 2x495MB if materialized -> fused so it never hits HBM.
//
// LDS holds theta_dir dir-PAIR interleaved per pixel so each WMMA B-fragment
// is one aligned ds_load_b64 straight into an even VGPR pair (no repack movs):
//   lds[(dirpair*6 + k)*LROW + 2*pix + (dir&1)],  dirpair = dir>>1
// ---------------------------------------------------------------------------

typedef __attribute__((ext_vector_type(2))) float v2f;
typedef __attribute__((ext_vector_type(8))) float v8f;

#define HW_      16384
#define C_       630
#define CT_      40              // ceil(630/16)
#define DIRS_    64
#define NK_      6
#define ROWS_    (DIRS_ * NK_)   // 384 source rows (dir-major, k-minor)
#define DPAIRS_  32              // 64 dirs / 2
#define PIXT_    64              // pixels per workgroup
#define LROW_    144             // dwords per (dirpair,k) row: 128 data + 16 pad
                                 // pad=16 -> low/high half-waves hit disjoint
                                 // 32-bank windows on ds_load_b64

__global__ __launch_bounds__(128, 1)
void taylor_fused_kernel(const float* __restrict__ tg,   // theta_global
                         const float* __restrict__ td,   // theta_dir
                         const float* __restrict__ tvec,
                         const float* __restrict__ bvec,
                         const float* __restrict__ P,
                         float* __restrict__ out)
{
    __shared__ __align__(16) float lds[DPAIRS_ * NK_ * LROW_];   // 110592 B

    const int tid      = threadIdx.x;
    const int batch    = blockIdx.y;
    const int pix_base = blockIdx.x * PIXT_;

    // ---- Stage theta_dir tile, interleaving even/odd dir rows per pixel.
    {
        const float* src = td + (size_t)batch * ROWS_ * HW_ + pix_base;
        for (int i = tid; i < DPAIRS_ * NK_ * 16; i += 128) {   // 24 iters
            const int dp  = i / (NK_ * 16);
            const int k   = (i >> 4) % NK_;
            const int seg = i & 15;                  // 4-pixel chunk
            const float* pe = src + ((size_t)(2 * dp) * NK_ + k) * HW_ + 4 * seg;
            const float* po = pe + (size_t)NK_ * HW_;     // dir = 2*dp + 1
            float4 e = *(const float4*)pe;
            float4 o = *(const float4*)po;
            float* dst = &lds[(dp * NK_ + k) * LROW_ + 8 * seg];
            *(float4*)(dst + 0) = make_float4(e.x, o.x, e.y, o.y);
            *(float4*)(dst + 4) = make_float4(e.z, o.z, e.w, o.w);
        }
    }
    __syncthreads();

    const int lane   = tid & 31;
    const int wave   = tid >> 5;
    const int nloc   = lane & 15;     // N position within 16x16 tile
    const int hi     = lane >> 4;     // half-wave select
    const int pixloc = wave * 16 + nloc;
    const int pix    = pix_base + pixloc;

    // Per-pixel globals (broadcast over c).
    const float g_ln  = tg[((size_t)batch * 4 + 0) * HW_ + pix];
    const float g_r   = tg[((size_t)batch * 4 + 1) * HW_ + pix];
    const float g_s20 = tg[((size_t)batch * 4 + 2) * HW_ + pix];
    const float g_s30 = tg[((size_t)batch * 4 + 3) * HW_ + pix];

    for (int ct = 0; ct < CT_; ++ct) {
        // A-fragment source row (clamped for padded tail; masked at store).
        const int crow  = ct * 16 + nloc;
        const int crowc = crow < C_ ? crow : (C_ - 1);
        const float* Arow = P + (size_t)crowc * DIRS_ + 2 * hi;

        v8f acc[NK_] = {};   // 6 x 16x16 f32 accumulators (48 VGPRs)

        #pragma unroll 4
        for (int kk = 0; kk < 16; ++kk) {            // K = 64 in steps of 4
            // A 16x4: v0/v1 = K = 4kk + 2*hi + {0,1}, M = lane%16  (one b64)
            const v2f a = *(const v2f*)(Arow + 4 * kk);
            // B 4x16: dirpair = 2kk + hi -> dirs {4kk+2hi, 4kk+2hi+1}
            const float* brow = &lds[((2 * kk + hi) * NK_) * LROW_ + 2 * pixloc];
            #pragma unroll
            for (int k = 0; k < NK_; ++k) {
                const v2f bf = *(const v2f*)(brow + k * LROW_);  // one ds_load_b64
                acc[k] = __builtin_amdgcn_wmma_f32_16x16x4_f32(
                    false, a, false, bf, (short)0, acc[k], false, false);
            }
        }

        // ---- Epilogue: D layout -> c_local = r + 8*hi, n = lane%16.
        #pragma unroll
        for (int r = 0; r < 8; ++r) {
            const int c  = ct * 16 + r + 8 * hi;
            const int cc = c < C_ ? c : (C_ - 1);
            const float t  = tvec[cc];
            const float b  = bvec[cc];
            const float t2 = t * t,  b2 = b * b,  tb = t * b;
            const float t3 = t2 * t, b3 = b2 * b;
            const float t2b = t2 * b, tb2 = t * b2;

            const float Dch = acc[0][r];
            const float s11 = acc[1][r];
            const float s02 = acc[2][r];
            const float s21 = acc[3][r];
            const float s12 = acc[4][r];
            const float s03 = acc[5][r];

            float log_s = g_ln - g_r * t - Dch * b
                        + 0.5f * (g_s20 * t2 + 2.0f * s11 * tb + s02 * b2)
                        - (1.0f / 6.0f) * (g_s30 * t3 + 3.0f * s21 * t2b
                                           + 3.0f * s12 * tb2 + s03 * b3);
            const float val = __expf(log_s);
            if (c < C_)
                out[((size_t)batch * C_ + c) * HW_ + pix] = val;
        }
    }
}

extern "C" void kernel_launch(void* const* d_in, const int* in_sizes, int n_in,
                              void* d_out, int out_size, void* d_ws, size_t ws_size,
                              hipStream_t stream) {
    const float* theta_global = (const float*)d_in[0];
    const float* theta_dir    = (const float*)d_in[1];
    const float* t_vec        = (const float*)d_in[2];
    const float* b_vec        = (const float*)d_in[3];
    const float* P            = (const float*)d_in[4];
    float* out = (float*)d_out;

    dim3 grid(HW_ / PIXT_, 2);   // 256 pixel-tiles x 2 batches
    taylor_fused_kernel<<<grid, 128, 0, stream>>>(
        theta_global, theta_dir, t_vec, b_vec, P, out);
}